// MLPKAN_71794673320509
// MI455X (gfx1250) — compile-verified
//
#include <hip/hip_runtime.h>
#include <stdint.h>

// ---------------------------------------------------------------------------
// MLP-KAN fused 2-layer forward for MI455X (gfx1250, wave32).
//   layer0: 64 -> 64, 4096 edge-subnets (1->2->2->1, relu)
//   layer1: 64 -> 16, 1024 edge-subnets
// One block = 32 batch rows, fully fused (hidden stays in LDS).
// CDNA5 paths: tensor_load_to_lds (TDM) for layer-1 params,
// s_wait_tensorcnt fence, v_wmma_f32_16x16x4_f32 reduction epilogue.
// ---------------------------------------------------------------------------

typedef float v2f __attribute__((ext_vector_type(2)));
typedef float v8f __attribute__((ext_vector_type(8)));
typedef unsigned int u32x4 __attribute__((ext_vector_type(4)));
typedef int i32x4 __attribute__((ext_vector_type(4)));
typedef int i32x8 __attribute__((ext_vector_type(8)));

#define BATCH    16384
#define IN0      64
#define OUT0     64
#define IN1      64
#define OUT1     16
#define L0_EDGES (IN0 * OUT0)   // 4096
#define L1_EDGES (IN1 * OUT1)   // 1024
#define BT       32             // batch rows per block

// d_ws layout (floats):
//   [0            .. 3*4096*4)   l0 packed params: q0[4096], q1[4096], q2[4096] (float4 each)
//   [49152        .. 49152+3*1024*4) l1 packed params: q0[1024], q1[1024], q2[1024]
//   [61440 .. 61504)  bias0[64]  = sum_i l0_b3[i*64+o]
//   [61504 .. 61520)  bias1[16]  = sum_i l1_b3[i*16+o]
#define WS_L1_FOFF   49152
#define WS_BIAS0     61440
#define WS_BIAS1     61504

// q0 = (W1[0], W1[1], b1[0], b1[1])
// q1 = (W2[0][0], W2[0][1], W2[1][0], W2[1][1])
// q2 = (b2[0], b2[1], W3[0], W3[1])

__global__ __launch_bounds__(256) void kan_prep_pad(
    const float* W1_0, const float* b1_0, const float* W2_0, const float* b2_0, const float* W3_0,
    const float* W1_1, const float* b1_1, const float* W2_1, const float* b2_1, const float* W3_1,
    float4* ws4)
{
    int e = blockIdx.x * blockDim.x + threadIdx.x;
    if (e < L0_EDGES) {
        ws4[e]                = make_float4(W1_0[2*e], W1_0[2*e+1], b1_0[2*e], b1_0[2*e+1]);
        ws4[L0_EDGES + e]     = make_float4(W2_0[4*e], W2_0[4*e+1], W2_0[4*e+2], W2_0[4*e+3]);
        ws4[2*L0_EDGES + e]   = make_float4(b2_0[2*e], b2_0[2*e+1], W3_0[2*e], W3_0[2*e+1]);
    } else if (e < L0_EDGES + L1_EDGES) {
        int n = e - L0_EDGES;
        const int base = 3 * L0_EDGES;   // float4 offset of l1 blob
        ws4[base + n]              = make_float4(W1_1[2*n], W1_1[2*n+1], b1_1[2*n], b1_1[2*n+1]);
        ws4[base + L1_EDGES + n]   = make_float4(W2_1[4*n], W2_1[4*n+1], W2_1[4*n+2], W2_1[4*n+3]);
        ws4[base + 2*L1_EDGES + n] = make_float4(b2_1[2*n], b2_1[2*n+1], W3_1[2*n], W3_1[2*n+1]);
    }
}

__global__ __launch_bounds__(128) void kan_prep_bias(
    const float* b3_0, const float* b3_1, float* wsF)
{
    int t = threadIdx.x;
    if (t < OUT0) {
        float s = 0.f;
        for (int i = 0; i < IN0; ++i) s += b3_0[i * OUT0 + t];
        wsF[WS_BIAS0 + t] = s;
    } else if (t < OUT0 + OUT1) {
        int o = t - OUT0;
        float s = 0.f;
        for (int i = 0; i < IN1; ++i) s += b3_1[i * OUT1 + o];
        wsF[WS_BIAS1 + o] = s;
    }
}

__device__ __forceinline__ float kan_edge(float xv, float4 q0, float4 q1, float4 q2, float acc)
{
    float u0 = fmaxf(fmaf(q0.x, xv, q0.z), 0.f);
    float u1 = fmaxf(fmaf(q0.y, xv, q0.w), 0.f);
    float v0 = fmaxf(fmaf(q1.y, u1, fmaf(q1.x, u0, q2.x)), 0.f);
    float v1 = fmaxf(fmaf(q1.w, u1, fmaf(q1.z, u0, q2.y)), 0.f);
    acc = fmaf(q2.z, v0, acc);
    acc = fmaf(q2.w, v1, acc);
    return acc;
}

__global__ __launch_bounds__(256) void kan_fused(
    const float* __restrict__ x, const float* __restrict__ wsF, float* __restrict__ out)
{
    __shared__ float4 l1qs[3 * L1_EDGES];     // 48 KB, filled by TDM
    __shared__ float  xs[BT][IN0];            // 8 KB  staged input tile
    __shared__ float  hbuf[BT][OUT0];         // 8 KB  hidden activations
    __shared__ float  part[BT][OUT1 * 4 + 1]; // ~8 KB layer-1 chunk partials (padded row)

    const int tid  = threadIdx.x;
    const int bblk = blockIdx.x * BT;

    // ---- Kick off async TDM copy of the 48 KB layer-1 param blob into LDS
    //      (wave 0 only; overlaps with all of layer-0 compute). -------------
#if defined(__has_builtin) && __has_builtin(__builtin_amdgcn_tensor_load_to_lds)
    if (tid < 32) {
        unsigned lds_off = (unsigned)__builtin_amdgcn_readfirstlane(
            (int)(uint32_t)(uintptr_t)&l1qs[0]);
        unsigned long long ga = (unsigned long long)(uintptr_t)(wsF + WS_L1_FOFF);
        unsigned galo = (unsigned)__builtin_amdgcn_readfirstlane((int)(unsigned)ga);
        unsigned gahi = (unsigned)__builtin_amdgcn_readfirstlane((int)(unsigned)(ga >> 32));
        // D# group0: count=1 | lds_addr | global_addr[56:0] | type=2
        u32x4 g0 = { 1u, lds_off, galo, (gahi & 0x01FFFFFFu) | 0x80000000u };
        // 1-D tensor: data_size=8B, dim0 = tile0 = 48KB/8 = 6144, dim1=1
        const unsigned TD0 = (3u * L1_EDGES * 16u) / 8u;  // 6144
        i32x8 g1 = { (int)(3u << 16),          // data_size=3 (8B)
                     (int)(TD0 << 16),          // tensor_dim0 lo16
                     (int)(1u << 16),           // tensor_dim0 hi16=0 | tensor_dim1 lo16=1
                     (int)(TD0 << 16),          // tensor_dim1 hi16=0 | tile_dim0
                     0,                         // tile_dim1=0 (unused), tile_dim2=0
                     (int)TD0,                  // tensor_dim0_stride lo32
                     0, 0 };
        i32x4 g2 = { 0, 0, 0, 0 };
        i32x4 g3 = { 0, 0, 0, 0 };
        i32x8 g4 = { 0, 0, 0, 0, 0, 0, 0, 0 }; // extra group (unused, 1-D tensor)
        __builtin_amdgcn_tensor_load_to_lds(g0, g1, g2, g3, g4, 0);
    }
#else
    { // fallback: cooperative LDS fill
        const float4* src = (const float4*)(wsF + WS_L1_FOFF);
        for (int k = tid; k < 3 * L1_EDGES; k += 256) l1qs[k] = src[k];
    }
#endif

    // ---- Stage x tile (coalesced b128) ------------------------------------
    {
        const float4* x4 = (const float4*)(x + (size_t)bblk * IN0);
        float4* xsv = (float4*)&xs[0][0];
        for (int k = tid; k < BT * IN0 / 4; k += 256) xsv[k] = x4[k];
    }
    __syncthreads();

    // ---- Layer 0: thread = (o, batch-group of 8); params stream from L2 ---
    {
        const int o  = tid & 63;
        const int bg = tid >> 6;
        const float4* q0t = (const float4*)wsF;
        const float4* q1t = q0t + L0_EDGES;
        const float4* q2t = q0t + 2 * L0_EDGES;
        const float bias_o = wsF[WS_BIAS0 + o];
        float acc[8] = {0.f,0.f,0.f,0.f,0.f,0.f,0.f,0.f};
        for (int i = 0; i < IN0; ++i) {
            const int n0 = i * OUT0 + o;
            const float4 q0 = q0t[n0], q1 = q1t[n0], q2 = q2t[n0];
            #pragma unroll
            for (int j = 0; j < 8; ++j)
                acc[j] = kan_edge(xs[bg * 8 + j][i], q0, q1, q2, acc[j]);
        }
        #pragma unroll
        for (int j = 0; j < 8; ++j)
            hbuf[bg * 8 + j][o] = acc[j] + bias_o;
    }

    // ---- Fence TDM before anyone reads l1qs -------------------------------
#if defined(__has_builtin) && __has_builtin(__builtin_amdgcn_s_wait_tensorcnt)
    if (tid < 32) __builtin_amdgcn_s_wait_tensorcnt(0);
#endif
    __syncthreads();

    // ---- Layer 1: thread = (o1, i-chunk of 16, batch-group of 8) ----------
    {
        const int o1 = tid & 15;
        const int ck = (tid >> 4) & 3;
        const int bg = tid >> 6;
        const float4* q0t = &l1qs[0];
        const float4* q1t = &l1qs[L1_EDGES];
        const float4* q2t = &l1qs[2 * L1_EDGES];
        float acc[8] = {0.f,0.f,0.f,0.f,0.f,0.f,0.f,0.f};
        for (int ii = 0; ii < 16; ++ii) {
            const int i  = ck * 16 + ii;
            const int n1 = i * OUT1 + o1;
            const float4 q0 = q0t[n1], q1 = q1t[n1], q2 = q2t[n1];
            #pragma unroll
            for (int j = 0; j < 8; ++j)
                acc[j] = kan_edge(hbuf[bg * 8 + j][i], q0, q1, q2, acc[j]);
        }
        #pragma unroll
        for (int j = 0; j < 8; ++j)
            part[bg * 8 + j][o1 * 4 + ck] = acc[j];
    }
    __syncthreads();

    // ---- WMMA epilogue: exact f32 K=4 chunk reduction via ones-matrix -----
    // D = A(16x4 of 1.0) x B(4x16 partials)  =>  D[m,n] = sum_k part[b=n][o1][k]
    {
        const int lane  = tid & 31;
        const int wid   = tid >> 5;
        const int n     = lane & 15;
        const int khalf = lane >> 4;   // VGPR0 carries K=0/2, VGPR1 carries K=1/3
        #pragma unroll
        for (int it = 0; it < 4; ++it) {
            const int task = wid * 4 + it;      // 8 waves x 4 = 32 tasks
            const int o1   = task & 15;
            const int half = task >> 4;
            const int b    = half * 16 + n;
            v2f av; av.x = 1.0f; av.y = 1.0f;
            v2f bv;
            bv.x = part[b][o1 * 4 + 2 * khalf];
            bv.y = part[b][o1 * 4 + 2 * khalf + 1];
            v8f c = {0.f,0.f,0.f,0.f,0.f,0.f,0.f,0.f};
            v8f d;
#if defined(__has_builtin) && __has_builtin(__builtin_amdgcn_wmma_f32_16x16x4_f32)
            d = __builtin_amdgcn_wmma_f32_16x16x4_f32(
                    false, av, false, bv, (short)0, c, false, false);
#else
            asm volatile("v_wmma_f32_16x16x4_f32 %0, %1, %2, %3"
                         : "=v"(d) : "v"(av), "v"(bv), "v"(c));
#endif
            if (lane < 16) {
                // D VGPR0, lanes 0..15 = row M=0, col N=lane
                out[(size_t)(bblk + half * 16 + lane) * OUT1 + o1] =
                    d[0] + wsF[WS_BIAS1 + o1];
            }
        }
    }
}

extern "C" void kernel_launch(void* const* d_in, const int* in_sizes, int n_in,
                              void* d_out, int out_size, void* d_ws, size_t ws_size,
                              hipStream_t stream)
{
    const float* x     = (const float*)d_in[0];
    const float* l0_W1 = (const float*)d_in[1];
    const float* l0_b1 = (const float*)d_in[2];
    const float* l0_W2 = (const float*)d_in[3];
    const float* l0_b2 = (const float*)d_in[4];
    const float* l0_W3 = (const float*)d_in[5];
    const float* l0_b3 = (const float*)d_in[6];
    const float* l1_W1 = (const float*)d_in[7];
    const float* l1_b1 = (const float*)d_in[8];
    const float* l1_W2 = (const float*)d_in[9];
    const float* l1_b2 = (const float*)d_in[10];
    const float* l1_W3 = (const float*)d_in[11];
    const float* l1_b3 = (const float*)d_in[12];

    float*  wsF = (float*)d_ws;
    float4* ws4 = (float4*)d_ws;
    float*  out = (float*)d_out;

    // Pack per-edge params into 3xfloat4 SoA tables (coalesced/aligned loads).
    kan_prep_pad<<<(L0_EDGES + L1_EDGES + 255) / 256, 256, 0, stream>>>(
        l0_W1, l0_b1, l0_W2, l0_b2, l0_W3,
        l1_W1, l1_b1, l1_W2, l1_b2, l1_W3, ws4);
    // Precompute column bias sums (removes a load+add per edge).
    kan_prep_bias<<<1, 128, 0, stream>>>(l0_b3, l1_b3, wsF);

    // Fused 2-layer KAN: 512 blocks x 256 threads, 32 batch rows per block.
    kan_fused<<<BATCH / BT, 256, 0, stream>>>(x, wsF, out);
    (void)in_sizes; (void)n_in; (void)out_size; (void)ws_size;
}